// APPNP_Conv_78529182040079
// MI455X (gfx1250) — compile-verified
//
#include <hip/hip_runtime.h>
#include <cstdint>
#include <cstddef>

#define D_FEAT 128
#define ALPHA_F 0.1f
#define TILE 32            // edges per wave batch (one b32 async op per lane per array)
#define WPB 8              // waves per block (256 threads = 8 wave32)
#define BLOCK (WPB * 32)

// --- CDNA5 async global->LDS copy (ASYNCcnt-tracked), inline asm per ISA 15.18.3 op 96.
// vdst = per-lane LDS byte address, vaddr = per-lane 64-bit global address, GV mode.
__device__ __forceinline__ void async_copy_b32(uint32_t lds_addr, const void* gptr) {
  uint64_t ga = (uint64_t)(uintptr_t)gptr;
  asm volatile("global_load_async_to_lds_b32 %0, %1, off"
               :: "v"(lds_addr), "v"(ga)
               : "memory");
}

// Low 32 bits of a generic pointer to a __shared__ object = LDS byte offset
// (hardware adds the wave's LDS_BASE; see ISA 10.2 aperture mapping).
__device__ __forceinline__ uint32_t lds_addr_of(const void* p) {
  return (uint32_t)(uintptr_t)p;
}

// Scalar (SGPR) broadcasts from a per-lane register: v_readlane_b32.
__device__ __forceinline__ int readlane_i(int v, int lane) {
  return __builtin_amdgcn_readlane(v, lane);
}
__device__ __forceinline__ float readlane_f(float v, int lane) {
  return __int_as_float(__builtin_amdgcn_readlane(__float_as_int(v), lane));
}

// One edge: (r, c, v) are wave-uniform scalars (SGPR); each lane owns 16
// contiguous bytes of the 512 B feature row -> saddr+voffset addressing.
// 4 hardware f32 atomics into the L2-resident accumulator.
__device__ __forceinline__ void process_edge(int r, int c, float v,
                                             const float* __restrict__ h,
                                             float* __restrict__ out, int lane) {
  const float4 hv = *reinterpret_cast<const float4*>(
      h + (uint32_t)c * (uint32_t)D_FEAT + (uint32_t)(lane * 4));
  float* op = out + (uint32_t)r * (uint32_t)D_FEAT + (uint32_t)(lane * 4);
  unsafeAtomicAdd(op + 0, v * hv.x);
  unsafeAtomicAdd(op + 1, v * hv.y);
  unsafeAtomicAdd(op + 2, v * hv.z);
  unsafeAtomicAdd(op + 3, v * hv.w);
}

__global__ __launch_bounds__(BLOCK) void appnp_scatter(
    const int* __restrict__ erow, const int* __restrict__ ecol,
    const float* __restrict__ eval, const float* __restrict__ h,
    float* __restrict__ out, int nEdges)
{
  // Per-wave private double buffers for the async edge stream.
  __shared__ int   sRow[WPB][2][TILE];
  __shared__ int   sCol[WPB][2][TILE];
  __shared__ float sVal[WPB][2][TILE];

  const int lane = threadIdx.x & 31;
  const int wv   = threadIdx.x >> 5;
  const int gw   = blockIdx.x * WPB + wv;       // global wave id
  const int nw   = gridDim.x * WPB;             // total waves
  const int nTiles = (nEdges + TILE - 1) / TILE;

  auto issue = [&](int t, int b) {
    int e = t * TILE + lane;
    if (e < nEdges) {
      async_copy_b32(lds_addr_of(&sRow[wv][b][lane]), &erow[e]);
      async_copy_b32(lds_addr_of(&sCol[wv][b][lane]), &ecol[e]);
      async_copy_b32(lds_addr_of(&sVal[wv][b][lane]), &eval[e]);
    }
  };

  if (gw >= nTiles) return;

  issue(gw, 0);                                  // prime buffer 0 (3 async ops)
  int buf = 0;
  const float scale = 1.0f - ALPHA_F;

  for (int t = gw; t < nTiles; t += nw) {
    int tn = t + nw;                             // wave-uniform
    if (tn < nTiles) {
      issue(tn, buf ^ 1);                        // 3 more in flight (total <= 6)
      // Async loads complete in order: <=3 outstanding => previous 3 landed.
      asm volatile("s_wait_asynccnt 3" ::: "memory");
    } else {
      asm volatile("s_wait_asynccnt 0" ::: "memory");
    }

    // Each lane pulls its own edge triple once (3 ds_load_b32 per tile);
    // per-edge broadcast below is v_readlane with immediate lane -> SGPRs.
    int   rreg = sRow[wv][buf][lane];
    int   creg = sCol[wv][buf][lane];
    float vreg = sVal[wv][buf][lane] * scale;

    // Warm all 32 gather rows for this tile: each lane prefetches its own
    // edge's full 512 B feature row (4 x 128 B lines) -> 4 global_prefetch_b8.
    {
      const float* hp = h + (uint32_t)creg * (uint32_t)D_FEAT;
      __builtin_prefetch(hp + 0,  0, 0);
      __builtin_prefetch(hp + 32, 0, 0);
      __builtin_prefetch(hp + 64, 0, 0);
      __builtin_prefetch(hp + 96, 0, 0);
    }

    int cnt = min(TILE, nEdges - t * TILE);
    if (cnt == TILE) {
#pragma unroll
      for (int i = 0; i < TILE; ++i) {           // full unroll: lane index is imm
        int   r = readlane_i(rreg, i);
        int   c = readlane_i(creg, i);
        float v = readlane_f(vreg, i);
        process_edge(r, c, v, h, out, lane);
      }
    } else {
      for (int i = 0; i < cnt; ++i) {
        int   r = readlane_i(rreg, i);
        int   c = readlane_i(creg, i);
        float v = readlane_f(vreg, i);
        process_edge(r, c, v, h, out, lane);
      }
    }
    buf ^= 1;
  }
}

__global__ __launch_bounds__(256) void appnp_init(const float4* __restrict__ h0,
                                                  float4* __restrict__ out, int n4) {
  int stride = gridDim.x * blockDim.x;
  for (int i = blockIdx.x * blockDim.x + threadIdx.x; i < n4; i += stride) {
    float4 a = h0[i];
    out[i] = make_float4(ALPHA_F * a.x, ALPHA_F * a.y, ALPHA_F * a.z, ALPHA_F * a.w);
  }
}

extern "C" void kernel_launch(void* const* d_in, const int* in_sizes, int n_in,
                              void* d_out, int out_size, void* d_ws, size_t ws_size,
                              hipStream_t stream) {
  const int*   erow = (const int*)d_in[0];
  const int*   ecol = (const int*)d_in[1];
  const float* eval = (const float*)d_in[2];
  const float* h    = (const float*)d_in[3];
  const float* h0   = (const float*)d_in[4];
  float* out = (float*)d_out;

  const int nEdges = in_sizes[0];
  const int n4 = out_size / 4;                  // N*D floats -> float4 count

  // out = ALPHA * h0 (re-initialized every call; scatter kernel accumulates on top)
  appnp_init<<<2048, 256, 0, stream>>>(
      (const float4*)h0, (float4*)out, n4);

  int nTiles = (nEdges + TILE - 1) / TILE;
  int blocks = (nTiles + WPB - 1) / WPB;
  if (blocks > 2048) blocks = 2048;             // ~16K waves, ~6 tiles each (double-buffered)
  appnp_scatter<<<blocks, BLOCK, 0, stream>>>(erow, ecol, eval, h, out, nEdges);
}